// BiLSTM_10548439679533
// MI455X (gfx1250) — compile-verified
//
#include <hip/hip_runtime.h>
#include <math.h>

#define T_LEN 2048
#define EDIM  512
#define HDIM  512
#define GDIM  2048   // 4*H
#define NC    7
#define NEGV  (-10000.0f)

typedef __attribute__((ext_vector_type(2))) float v2f;
typedef __attribute__((ext_vector_type(8))) float v8f;

// async LDS-DMA builtin pointer types (from hipcc diagnostics):
//   b128: (int4 AS1*, int4 AS3*, imm, imm) ; b32: (int AS1*, int AS3*, imm, imm)
typedef __attribute__((ext_vector_type(4))) int i32x4;
typedef __attribute__((address_space(1))) i32x4 g_i32x4;
typedef __attribute__((address_space(3))) i32x4 l_i32x4;
typedef __attribute__((address_space(1))) int   g_i32;
typedef __attribute__((address_space(3))) int   l_i32;

#if __has_builtin(__builtin_amdgcn_global_load_async_to_lds_b128) && \
    __has_builtin(__builtin_amdgcn_global_load_async_to_lds_b32)
#define HAS_ASYNC_LDS 1
#else
#define HAS_ASYNC_LDS 0
#endif

__device__ __forceinline__ void wait_async0() {
#if __has_builtin(__builtin_amdgcn_s_wait_asynccnt)
    __builtin_amdgcn_s_wait_asynccnt(0);
#else
    asm volatile("s_wait_asynccnt 0" ::: "memory");
#endif
}

// ---------------------------------------------------------------- embedding
__global__ void gather_embed(const int* __restrict__ x, const float* __restrict__ emb,
                             float* __restrict__ xe) {
    int t = blockIdx.x;
    int row = x[t];
    const float4* src = (const float4*)(emb + (size_t)row * EDIM);
    float4* dst = (float4*)(xe + (size_t)t * EDIM);
    for (int i = threadIdx.x; i < EDIM / 4; i += blockDim.x) dst[i] = src[i];
}

// ------------------------------------------------- input projection (f32 WMMA)
// xz[t, n] = sum_k xe[src_t, k] * W[n, k] + b_ih[n] + b_hh[n]
// M = T (2048), N = 4H (2048), K = E (512). One 16x16 tile per wave,
// K-loop in steps of 4 using V_WMMA_F32_16X16X4_F32 (keeps full f32 precision).
__global__ void input_proj(const float* __restrict__ xe, const float* __restrict__ W,
                           const float* __restrict__ b_ih, const float* __restrict__ b_hh,
                           float* __restrict__ xz, int reversed) {
    const int wave = threadIdx.x >> 5;
    const int lane = threadIdx.x & 31;
    const int mTile = blockIdx.x >> 4;                 // 128 M tiles
    const int nTile = ((blockIdx.x & 15) << 3) + wave; // 128 N tiles
    const int m0 = mTile * 16, n0 = nTile * 16;

    const int mi = lane & 15;   // M for A-frag, N for B/D-frag
    const int kh = lane >> 4;   // K-half select (K offset 2*kh)
    const int n  = n0 + mi;
    const int mRow = m0 + mi;
    const int xeRow = reversed ? (T_LEN - 1 - mRow) : mRow;

    const float* aPtr = xe + (size_t)xeRow * EDIM + 2 * kh;
    const float* bPtr = W  + (size_t)n     * EDIM + 2 * kh;

    v8f acc = {};
    for (int k = 0; k < EDIM; k += 4) {
        v2f a = *(const v2f*)(aPtr + k);
        v2f b = *(const v2f*)(bPtr + k);
        acc = __builtin_amdgcn_wmma_f32_16x16x4_f32(
            false, a, false, b, (short)0, acc, false, false);
    }
    const float bias = b_ih[n] + b_hh[n];
    #pragma unroll
    for (int v = 0; v < 8; ++v) {
        int mr = m0 + 8 * kh + v;                      // D layout: M = v + 8*(lane>>4)
        xz[(size_t)mr * GDIM + n] = acc[v] + bias;
    }
}

// ---------------------------------------------------------------- init state
__global__ void init_state(const float* __restrict__ h0, float* __restrict__ h_glob,
                           int* __restrict__ bar) {
    int i = threadIdx.x;
    if (i < 2 * HDIM) h_glob[i] = h0[i];
    if (i < 4) bar[i] = 0;
}

__device__ __forceinline__ float sigmoidf_(float x) { return 1.0f / (1.0f + __expf(-x)); }

// ----------------------------------------------------- sequential recurrence
// 32 blocks: 16 per direction. Block jj of a direction owns h indices
// [32*jj, 32*jj+32) -> 128 gate rows (i,f,g,o) of W_hh, staged once into LDS
// via async LDS-DMA (256 KB slice; only possible with CDNA5's 320 KB LDS/WGP).
// Per step: LDS matvec -> gates -> (c,h) update -> publish 32 h's ->
// device-scope generation barrier -> async h broadcast reload into LDS.
__global__ void lstm_recur(const float* __restrict__ xzf, const float* __restrict__ xzb,
                           const float* __restrict__ Whf, const float* __restrict__ Whb,
                           const float* __restrict__ c0, float* __restrict__ h_glob,
                           float* __restrict__ hf, float* __restrict__ hb,
                           int* __restrict__ bar) {
    extern __shared__ float smem[];
    float* Wsl  = smem;                 // 128 * 512 f32 = 256 KB
    float* hcur = smem + 128 * HDIM;    // 512 f32
    float* gat  = hcur + HDIM;          // 128 f32

    const int tid   = threadIdx.x;
    const int dir   = blockIdx.x >> 4;
    const int jj    = blockIdx.x & 15;
    const int hBase = jj * 32;

    const float* W  = dir ? Whb : Whf;
    const float* xz = dir ? xzb : xzf;
    float*       hs = dir ? hb  : hf;
    int* cnt = bar + dir;
    int* gen = bar + 2 + dir;

    // Stage W_hh slice: LDS row r (0..127) <- global row (r>>5)*H + hBase + (r&31)
#if HAS_ASYNC_LDS
    for (int r = 0; r < 128; ++r) {
        int gRow = (r >> 5) * HDIM + hBase + (r & 31);
        const float* src = W + (size_t)gRow * HDIM;
        float* dst = Wsl + (size_t)r * HDIM;
        for (int i = tid * 4; i < HDIM; i += blockDim.x * 4)
            __builtin_amdgcn_global_load_async_to_lds_b128(
                (g_i32x4*)(void*)(src + i), (l_i32x4*)(void*)(dst + i), 0, 0);
    }
    for (int i = tid; i < HDIM; i += blockDim.x)
        __builtin_amdgcn_global_load_async_to_lds_b32(
            (g_i32*)(void*)(h_glob + dir * HDIM + i), (l_i32*)(void*)(hcur + i), 0, 0);
    wait_async0();
#else
    for (int r = 0; r < 128; ++r) {
        int gRow = (r >> 5) * HDIM + hBase + (r & 31);
        const float4* src = (const float4*)(W + (size_t)gRow * HDIM);
        float4* dst = (float4*)(Wsl + (size_t)r * HDIM);
        for (int i = tid; i < HDIM / 4; i += blockDim.x) dst[i] = src[i];
    }
    for (int i = tid; i < HDIM; i += blockDim.x) hcur[i] = h_glob[dir * HDIM + i];
#endif
    float cval = (tid < 32) ? c0[dir * HDIM + hBase + tid] : 0.0f;
    __syncthreads();

    const int r    = tid >> 1;      // gate row 0..127
    const int half = tid & 1;       // K half (0..255 / 256..511)
    const float4* wrow = (const float4*)(Wsl + (size_t)r * HDIM + half * 256);
    const float4* hvec = (const float4*)(hcur + half * 256);
    const int col = (r >> 5) * HDIM + hBase + (r & 31);   // column in xz [4H]

    for (int t = 0; t < T_LEN; ++t) {
        if (t + 1 < T_LEN)   // hide next step's only global read on the critical path
            __builtin_prefetch(xz + (size_t)(t + 1) * GDIM + col, 0, 1);

        float s = 0.0f;
        #pragma unroll 8
        for (int i = 0; i < 64; ++i) {
            float4 w = wrow[i];
            float4 h = hvec[i];
            s += w.x * h.x + w.y * h.y + w.z * h.z + w.w * h.w;
        }
        s += __shfl_xor(s, 1);
        if (half == 0) gat[r] = s + xz[(size_t)t * GDIM + col];
        __syncthreads();

        if (tid < 32) {
            float zi = gat[tid], zf = gat[32 + tid], zg = gat[64 + tid], zo = gat[96 + tid];
            float ig = sigmoidf_(zi), fg = sigmoidf_(zf);
            float gg = tanhf(zg),     og = sigmoidf_(zo);
            cval = fg * cval + ig * gg;
            float hv = og * tanhf(cval);
            h_glob[dir * HDIM + hBase + tid] = hv;
            int outT = dir ? (T_LEN - 1 - t) : t;
            hs[(size_t)outT * HDIM + hBase + tid] = hv;
        }
        __threadfence();
        __syncthreads();
        if (tid == 0) {      // generation barrier across the 16 blocks of this dir
            int g = __atomic_load_n(gen, __ATOMIC_ACQUIRE);
            int a = __atomic_fetch_add(cnt, 1, __ATOMIC_ACQ_REL);
            if (a == 15) {
                __atomic_store_n(cnt, 0, __ATOMIC_RELAXED);
                __atomic_fetch_add(gen, 1, __ATOMIC_RELEASE);
            } else {
                while (__atomic_load_n(gen, __ATOMIC_ACQUIRE) == g)
                    __builtin_amdgcn_s_sleep(2);
            }
        }
        __syncthreads();
        __threadfence();
        // reload full h broadcast into LDS for next step's matvec
#if HAS_ASYNC_LDS
        for (int i = tid; i < HDIM; i += blockDim.x)
            __builtin_amdgcn_global_load_async_to_lds_b32(
                (g_i32*)(void*)(h_glob + dir * HDIM + i), (l_i32*)(void*)(hcur + i), 0, 0);
        wait_async0();
#else
        for (int i = tid; i < HDIM; i += blockDim.x) hcur[i] = h_glob[dir * HDIM + i];
#endif
        __syncthreads();
    }
}

// --------------------------------------------------------- output projection
__global__ void linear_out(const float* __restrict__ hf, const float* __restrict__ hb,
                           const float* __restrict__ Wlin, const float* __restrict__ blin,
                           float* __restrict__ feats) {
    const int t = blockIdx.x;
    const int c = threadIdx.x >> 5;
    const int lane = threadIdx.x & 31;
    if (c >= NC) return;
    float s = 0.0f;
    for (int k = lane; k < HDIM; k += 32) s += hf[(size_t)t * HDIM + k] * Wlin[c * 1024 + k];
    for (int k = lane; k < HDIM; k += 32) s += hb[(size_t)t * HDIM + k] * Wlin[c * 1024 + 512 + k];
    #pragma unroll
    for (int off = 16; off > 0; off >>= 1) s += __shfl_xor(s, off);
    if (lane == 0) feats[t * NC + c] = s + blin[c];
}

// -------------------------------------------------------- Viterbi (one wave32)
__global__ void viterbi(const float* __restrict__ feats, const float* __restrict__ trans,
                        float* __restrict__ out) {
    __shared__ short bp[T_LEN * NC];
    __shared__ float term_s[NC];
    const int lane = threadIdx.x;
    const int c = (lane < NC) ? lane : 0;

    float tr[NC];
    #pragma unroll
    for (int p = 0; p < NC; ++p) tr[p] = trans[c * NC + p];

    float fv = (lane == 5) ? 0.0f : NEGV;   // START=5
    for (int t = 0; t < T_LEN; ++t) {
        float best = tr[0] + __shfl(fv, 0);
        int bi = 0;
        #pragma unroll
        for (int p = 1; p < NC; ++p) {
            float v = tr[p] + __shfl(fv, p);
            if (v > best) { best = v; bi = p; }     // first-max wins (strict >)
        }
        fv = best + feats[t * NC + c];
        if (lane < NC) bp[t * NC + lane] = (short)bi;
    }
    if (lane < NC) term_s[lane] = fv + trans[6 * NC + lane];   // STOP=6
    __syncthreads();
    if (lane == 0) {
        int cur = bp[(T_LEN - 1) * NC + (NC - 1)];  // reference bug: bps[-1, C-1]
        out[0] = term_s[cur];
        out[T_LEN] = (float)cur;                    // best_path[T-1]
        for (int t = T_LEN - 1; t >= 1; --t) {
            cur = bp[t * NC + cur];
            out[t] = (float)cur;                    // best_path[t-1] at out[1+(t-1)]
        }
    }
}

// ---------------------------------------------------------------------------
extern "C" void kernel_launch(void* const* d_in, const int* in_sizes, int n_in,
                              void* d_out, int out_size, void* d_ws, size_t ws_size,
                              hipStream_t stream) {
    (void)in_sizes; (void)n_in; (void)out_size; (void)ws_size;
    const int*   x     = (const int*)d_in[0];
    const float* emb   = (const float*)d_in[1];
    const float* Wif   = (const float*)d_in[2];
    const float* Whf   = (const float*)d_in[3];
    const float* bif   = (const float*)d_in[4];
    const float* bhf   = (const float*)d_in[5];
    const float* Wib   = (const float*)d_in[6];
    const float* Whb   = (const float*)d_in[7];
    const float* bib   = (const float*)d_in[8];
    const float* bhb   = (const float*)d_in[9];
    const float* Wlin  = (const float*)d_in[10];
    const float* blin  = (const float*)d_in[11];
    const float* trans = (const float*)d_in[12];
    const float* h0    = (const float*)d_in[13];
    const float* c0    = (const float*)d_in[14];
    float* out = (float*)d_out;

    // workspace layout (floats)
    float* xe     = (float*)d_ws;
    float* xzf    = xe   + (size_t)T_LEN * EDIM;
    float* xzb    = xzf  + (size_t)T_LEN * GDIM;
    float* hf     = xzb  + (size_t)T_LEN * GDIM;
    float* hb     = hf   + (size_t)T_LEN * HDIM;
    float* feats  = hb   + (size_t)T_LEN * HDIM;
    float* h_glob = feats + (size_t)T_LEN * NC;
    int*   bar    = (int*)(h_glob + 2 * HDIM);

    const size_t smem_recur = (size_t)(128 * HDIM + HDIM + 128) * sizeof(float);
    (void)hipFuncSetAttribute((const void*)lstm_recur,
                              hipFuncAttributeMaxDynamicSharedMemorySize,
                              (int)smem_recur);

    gather_embed<<<T_LEN, 256, 0, stream>>>(x, emb, xe);
    input_proj  <<<2048, 256, 0, stream>>>(xe, Wif, bif, bhf, xzf, 0);
    input_proj  <<<2048, 256, 0, stream>>>(xe, Wib, bib, bhb, xzb, 1);
    init_state  <<<1, 1024, 0, stream>>>(h0, h_glob, bar);
    lstm_recur  <<<32, 256, smem_recur, stream>>>(xzf, xzb, Whf, Whb, c0, h_glob, hf, hb, bar);
    linear_out  <<<T_LEN, 256, 0, stream>>>(hf, hb, Wlin, blin, feats);
    viterbi     <<<1, 32, 0, stream>>>(feats, trans, out);
}